// DOMPooling_15985868276240
// MI455X (gfx1250) — compile-verified
//
#include <hip/hip_runtime.h>

// Problem constants (fixed by the reference file)
#define TOTAL_PULSES 4194304
#define EMBED_DIM    64
#define NUM_DOMS     262144

typedef float v2f __attribute__((ext_vector_type(2)));
typedef float v8f __attribute__((ext_vector_type(8)));

// ---------------------------------------------------------------------------
// Ordered-int float atomic max: guarantees native global_atomic_max/min
// (no CAS loop). Valid for all non-NaN IEEE floats; buffer initialized -inf.
// ---------------------------------------------------------------------------
__device__ __forceinline__ void atomicMaxFloat(float* addr, float v) {
    if (v >= 0.0f)
        atomicMax((int*)addr, __float_as_int(v));
    else
        atomicMin((unsigned int*)addr, __float_as_uint(v));
}

// ---------------------------------------------------------------------------
// Kernel 1: initialize workspace (must run every call; graph replays).
// ---------------------------------------------------------------------------
__global__ void dom_init_ws(float* __restrict__ sum, float* __restrict__ mx,
                            float* __restrict__ cnt) {
    int tid = blockIdx.x * blockDim.x + threadIdx.x;
    const int total = NUM_DOMS * EMBED_DIM;
    if (tid < total) {
        sum[tid] = 0.0f;
        mx[tid]  = __int_as_float(0xFF800000);  // -inf
    }
    if (tid < NUM_DOMS) cnt[tid] = 0.0f;
}

// ---------------------------------------------------------------------------
// Kernel 2: scatter sum / max / count. One thread per float4 (16 thr/pulse).
// Embedding read is perfectly coalesced b128 loads; atomics land in L2
// (129 MiB working set < 192 MB L2).
// ---------------------------------------------------------------------------
__global__ void dom_scatter(const float* __restrict__ emb,
                            const int*   __restrict__ idx,
                            float* __restrict__ sum, float* __restrict__ mx,
                            float* __restrict__ cnt) {
    int tid = blockIdx.x * blockDim.x + threadIdx.x;   // over TOTAL_PULSES*16
    int p = tid >> 4;          // pulse
    int q = tid & 15;          // float4 slot within the 64-dim embedding
    int d = idx[p];            // dom id
    float4 v = ((const float4*)emb)[tid];
    float* s = sum + d * EMBED_DIM + q * 4;
    float* m = mx  + d * EMBED_DIM + q * 4;
    atomicAdd(s + 0, v.x); atomicMaxFloat(m + 0, v.x);
    atomicAdd(s + 1, v.y); atomicMaxFloat(m + 1, v.y);
    atomicAdd(s + 2, v.z); atomicMaxFloat(m + 2, v.z);
    atomicAdd(s + 3, v.w); atomicMaxFloat(m + 3, v.w);
    if (q == 0) atomicAdd(&cnt[d], 1.0f);
}

// ---------------------------------------------------------------------------
// Kernel 3: fused finalize (mean = sum/max(cnt,1); empty max -> 0) + GEMM
// out = concat(mean,max) @ proj_w^T + bias   via V_WMMA_F32_16X16X4_F32.
// One wave32 per 16-row M-tile, 4 accumulators cover all 64 output cols.
// K loop split at the mean/max boundary (uniform in k) -> branch-free body.
// ---------------------------------------------------------------------------
__global__ void dom_pool_proj_wmma(const float* __restrict__ sum,
                                   const float* __restrict__ mx,
                                   const float* __restrict__ cnt,
                                   const float* __restrict__ w,     // (64,128) row-major
                                   const float* __restrict__ bias,  // (64,)
                                   float* __restrict__ out) {       // (NUM_DOMS,64)
    const int lane  = threadIdx.x & 31;
    const int wave  = threadIdx.x >> 5;
    const int mtile = blockIdx.x * 8 + wave;       // 0 .. NUM_DOMS/16 - 1
    const int hl    = lane >> 4;                   // lane half: K offset group
    const int lm    = lane & 15;

    // This lane supplies A rows M = lm (per WMMA f32 A layout)
    const int arow = mtile * 16 + lm;
    const float c     = cnt[arow];
    const float inv   = 1.0f / fmaxf(c, 1.0f);
    const bool  empty = (c == 0.0f);
    const float* srow = sum + arow * EMBED_DIM + hl * 2;
    const float* mrow = mx  + arow * EMBED_DIM + hl * 2;

    // B columns n = nt*16 + lm; proj_w row-major -> consecutive K = float2
    const float* w0 = w + (0 * 16 + lm) * 128 + hl * 2;
    const float* w1 = w + (1 * 16 + lm) * 128 + hl * 2;
    const float* w2 = w + (2 * 16 + lm) * 128 + hl * 2;
    const float* w3 = w + (3 * 16 + lm) * 128 + hl * 2;

    v8f acc0 = {}; v8f acc1 = {}; v8f acc2 = {}; v8f acc3 = {};
    const v2f inv2 = {inv, inv};
    const v2f zero2 = {0.0f, 0.0f};

    // ---- mean half: K = 0..63 (ka = k + hl*2 always < 64 here) ----
    #pragma unroll
    for (int k = 0; k < 64; k += 4) {
        v2f a = *(const v2f*)(srow + k) * inv2;          // v_pk_mul_f32
        v2f b0 = *(const v2f*)(w0 + k);
        v2f b1 = *(const v2f*)(w1 + k);
        v2f b2 = *(const v2f*)(w2 + k);
        v2f b3 = *(const v2f*)(w3 + k);
        acc0 = __builtin_amdgcn_wmma_f32_16x16x4_f32(false, a, false, b0, (short)0, acc0, false, false);
        acc1 = __builtin_amdgcn_wmma_f32_16x16x4_f32(false, a, false, b1, (short)0, acc1, false, false);
        acc2 = __builtin_amdgcn_wmma_f32_16x16x4_f32(false, a, false, b2, (short)0, acc2, false, false);
        acc3 = __builtin_amdgcn_wmma_f32_16x16x4_f32(false, a, false, b3, (short)0, acc3, false, false);
    }

    // ---- max half: K = 64..127 (empty segments contribute 0, not -inf) ----
    #pragma unroll
    for (int k = 0; k < 64; k += 4) {
        v2f mv = *(const v2f*)(mrow + k);
        v2f a  = empty ? zero2 : mv;                     // v_cndmask, no divergence
        v2f b0 = *(const v2f*)(w0 + 64 + k);
        v2f b1 = *(const v2f*)(w1 + 64 + k);
        v2f b2 = *(const v2f*)(w2 + 64 + k);
        v2f b3 = *(const v2f*)(w3 + 64 + k);
        acc0 = __builtin_amdgcn_wmma_f32_16x16x4_f32(false, a, false, b0, (short)0, acc0, false, false);
        acc1 = __builtin_amdgcn_wmma_f32_16x16x4_f32(false, a, false, b1, (short)0, acc1, false, false);
        acc2 = __builtin_amdgcn_wmma_f32_16x16x4_f32(false, a, false, b2, (short)0, acc2, false, false);
        acc3 = __builtin_amdgcn_wmma_f32_16x16x4_f32(false, a, false, b3, (short)0, acc3, false, false);
    }

    // D layout: VGPR r holds row (r + 8*hl), col lm (within each 16-col tile)
    const float bs0 = bias[ 0 + lm];
    const float bs1 = bias[16 + lm];
    const float bs2 = bias[32 + lm];
    const float bs3 = bias[48 + lm];
    #pragma unroll
    for (int r = 0; r < 8; ++r) {
        float* orow = out + (mtile * 16 + r + hl * 8) * EMBED_DIM;
        orow[ 0 + lm] = acc0[r] + bs0;
        orow[16 + lm] = acc1[r] + bs1;
        orow[32 + lm] = acc2[r] + bs2;
        orow[48 + lm] = acc3[r] + bs3;
    }
}

// ---------------------------------------------------------------------------
extern "C" void kernel_launch(void* const* d_in, const int* in_sizes, int n_in,
                              void* d_out, int out_size, void* d_ws, size_t ws_size,
                              hipStream_t stream) {
    const float* emb  = (const float*)d_in[0];   // (TOTAL_PULSES, 64) f32
    const int*   idx  = (const int*)  d_in[1];   // (TOTAL_PULSES,) int
    const float* w    = (const float*)d_in[3];   // (64, 128) f32
    const float* bias = (const float*)d_in[4];   // (64,) f32
    float*       out  = (float*)d_out;           // (NUM_DOMS, 64) f32

    float* sum = (float*)d_ws;                        //  64 MiB
    float* mx  = sum + (size_t)NUM_DOMS * EMBED_DIM;  //  64 MiB
    float* cnt = mx  + (size_t)NUM_DOMS * EMBED_DIM;  //   1 MiB

    {   // init
        int total = NUM_DOMS * EMBED_DIM;
        dom_init_ws<<<(total + 255) / 256, 256, 0, stream>>>(sum, mx, cnt);
    }
    {   // scatter
        int total = TOTAL_PULSES * 16;           // float4 granules
        dom_scatter<<<total / 256, 256, 0, stream>>>(emb, idx, sum, mx, cnt);
    }
    {   // fused finalize + projection GEMM (WMMA)
        int mtiles = NUM_DOMS / 16;              // 16384 waves
        dom_pool_proj_wmma<<<mtiles / 8, 256, 0, stream>>>(sum, mx, cnt, w, bias, out);
    }
}